// MiniStageGRU_22505628631515
// MI455X (gfx1250) — compile-verified
//
#include <hip/hip_runtime.h>
#include <math.h>

typedef __attribute__((ext_vector_type(2))) float v2f;
typedef __attribute__((ext_vector_type(8))) float v8f;

namespace {
constexpr int NB = 16;
constexpr int NT = 1000;
constexpr int NF = 257;
constexpr int FW = 284;   // padded freq stride in LDS (covers [-6, 277])
constexpr int PADF = 6;   // left halo pad

// LDS float-offset map
constexpr int XBASE  = 0;                   // x : 4ch x 4 rows (t-3..t)
constexpr int HBASE  = XBASE  + 4*4*FW;     // h : 8ch x 3 rows (t-2..t)
constexpr int XPBASE = HBASE  + 8*3*FW;     // xp: 8ch x 3 rows (t-2..t)
constexpr int RHBASE = XPBASE + 8*3*FW;     // r*h: 8ch x 2 rows (t-1..t)
constexpr int ZBASE  = RHBASE + 8*2*FW;     // z : 8ch x 1 row (t)
constexpr int STOT   = ZBASE  + 8*FW;
constexpr int ZEROTO = RHBASE + 8*2*FW;     // zero-init x..rh (halo correctness)

// K layout: per (src, ic, dt) a padded group of 6 taps (df 0..4 valid, df==5 dummy w=0)
// xp conv : K = 4ic * 2dt * 6  = 48  -> 12 chunks of 4
// zr/n    : K = 2src * 8ic * 2dt * 6 = 192 -> 48 chunks of 4
}

__device__ __forceinline__ float fast_sigmoid(float v) {
  return 1.f / (1.f + __expf(-v));
}
__device__ __forceinline__ float fast_tanh(float v) {
  // branchless; saturates correctly for |v| large (exp -> inf/0)
  return 1.f - 2.f / (__expf(2.f * v) + 1.f);
}
__device__ __forceinline__ float fast_elu(float v) {
  float e = __expf(fminf(v, 0.f)) - 1.f;   // no overflow, no branch
  return (v > 0.f) ? v : e;
}

__global__ __launch_bounds__(256)
void ministage_gru_fused(
    const float* __restrict__ xg, const float* __restrict__ hg,
    const float* __restrict__ pre_w, const float* __restrict__ pre_b,
    const float* __restrict__ xz_w, const float* __restrict__ xz_b,
    const float* __restrict__ xr_w, const float* __restrict__ xr_b,
    const float* __restrict__ xn_w, const float* __restrict__ xn_b,
    const float* __restrict__ hz_w, const float* __restrict__ hz_b,
    const float* __restrict__ hr_w, const float* __restrict__ hr_b,
    const float* __restrict__ hn_w, const float* __restrict__ hn_b,
    float* __restrict__ out)
{
  __shared__ float S[STOT];
  __shared__ v2f Bxp2[12*32];    // pre-conv weights, per-lane K pairs (ds_load_b64)
  __shared__ v2f Bzr2[48*32];    // cols 0-7: z (xz|hz), cols 8-15: r (xr|hr)
  __shared__ v2f Bn2 [48*32];    // cols 0-7: n (xn|hn), cols 8-15: zero
  __shared__ int OFFxp[24], OFFzr[96], OFFn[96];   // per even-K pair LDS base
  __shared__ float biasXP[16], biasZR[16], biasN[16];

  const int t    = blockIdx.x;
  const int b    = blockIdx.y;
  const int tid  = threadIdx.x;
  const int lane = tid & 31;
  const int wave = tid >> 5;
  const int g    = lane >> 4;   // half-wave group
  const int m    = lane & 15;   // A row (freq) / D column (out-chan)

  // ---- init: zero halo regions; build weight/offset/bias tables ----
  for (int i = tid; i < ZEROTO; i += 256) S[i] = 0.f;

  for (int e = tid; e < 12*64; e += 256) {           // Bxp
    int ch = e >> 6, rem = e & 63, ln = rem >> 1, v = rem & 1;
    int k = 4*ch + 2*(ln >> 4) + v;                  // 0..47
    int n = ln & 15;
    int ic = k / 12, r = k % 12, dt = r / 6, df = r % 6;
    float w = 0.f;
    if (n < 8 && df < 5) w = pre_w[((n*4 + ic)*2 + dt)*5 + df];
    Bxp2[ch*32 + ln][v] = w;
  }
  for (int e = tid; e < 48*64; e += 256) {           // Bzr / Bn
    int ch = e >> 6, rem = e & 63, ln = rem >> 1, v = rem & 1;
    int k = 4*ch + 2*(ln >> 4) + v;                  // 0..191
    int n = ln & 15, oc = n & 7;
    int kk = (k < 96) ? k : k - 96;
    int ic = kk / 12, r = kk % 12, dt = r / 6, df = r % 6;
    float wz = 0.f, wn = 0.f;
    if (df < 5) {
      int widx = ((oc*8 + ic)*2 + dt)*5 + df;
      const float* wzp = (k < 96) ? ((n < 8) ? xz_w : xr_w)
                                  : ((n < 8) ? hz_w : hr_w);
      wz = wzp[widx];
      if (n < 8) wn = ((k < 96) ? xn_w : hn_w)[widx];
    }
    Bzr2[ch*32 + ln][v] = wz;
    Bn2 [ch*32 + ln][v] = wn;
  }
  for (int p = tid; p < 24; p += 256) {
    int k = 2*p;
    int ic = k / 12, r = k % 12, dt = r / 6, df = r % 6;
    OFFxp[p] = XBASE + (ic*4 + dt)*FW + df;
  }
  for (int p = tid; p < 96; p += 256) {
    int k = 2*p;
    int kk = (k < 96) ? k : k - 96;
    int ic = kk / 12, r = kk % 12, dt = r / 6, df = r % 6;
    OFFzr[p] = ((k < 96) ? XPBASE : HBASE) + (ic*3 + dt)*FW + df;
    OFFn[p]  = (k < 96) ? (XPBASE + (ic*3 + 1 + dt)*FW + df)   // xp rows t-1..t
                        : (RHBASE + (ic*2 + dt)*FW + df);      // rh rows t-1..t
  }
  if (tid < 16) {
    int n = tid, oc = n & 7;
    biasXP[n] = (n < 8) ? pre_b[n] : 0.f;
    biasZR[n] = (n < 8) ? (xz_b[oc] + hz_b[oc]) : (xr_b[oc] + hr_b[oc]);
    biasN[n]  = (n < 8) ? (xn_b[oc] + hn_b[oc]) : 0.f;
  }
  __syncthreads();

  // ---- stage x (rows t-3..t) and h (rows t-2..t) into LDS ----
  for (int i = tid; i < 4*4*NF; i += 256) {
    int f = i % NF, r2 = i / NF, rw = r2 & 3, ic = r2 >> 2;
    int tr = t - 3 + rw;
    if (tr >= 0)
      S[XBASE + (ic*4 + rw)*FW + PADF + f] = xg[((b*4 + ic)*NT + tr)*NF + f];
  }
  for (int i = tid; i < 8*3*NF; i += 256) {
    int f = i % NF, r2 = i / NF, rw = r2 % 3, ic = r2 / 3;
    int tr = t - 2 + rw;
    if (tr >= 0)
      S[HBASE + (ic*3 + rw)*FW + PADF + f] = hg[((b*8 + ic)*NT + tr)*NF + f];
  }
  __syncthreads();

  // ---- phase 1: xp = elu(conv(x,pre)) at rows t-2..t, freq [-4,267] ----
  for (int task = wave; task < 3*17; task += 8) {
    int row = task / 17, tile = task % 17;
    int f0 = -4 + 16*tile;
    int ax = row*FW + PADF + f0 + m - 2;
    float bias = biasXP[m];
    v8f c;
#pragma unroll
    for (int j = 0; j < 8; ++j) c[j] = bias;
    for (int chk = 0; chk < 12; ++chk) {
      const float* p = &S[OFFxp[2*chk + g] + ax];
      v2f a; a.x = p[0]; a.y = p[1];
      v2f bb = Bxp2[chk*32 + lane];
      c = __builtin_amdgcn_wmma_f32_16x16x4_f32(false, a, false, bb, (short)0, c, false, false);
    }
    if (m < 8) {
      int base = XPBASE + (m*3 + row)*FW + PADF + f0 + 8*g;
#pragma unroll
      for (int j = 0; j < 8; ++j)
        S[base + j] = fast_elu(c[j]);
    }
  }
  __syncthreads();

  // ---- phase 2: z (row t) and r*h (rows t-1..t); K = xp ++ h, N: z|r ----
  for (int task = wave; task < 2*17; task += 8) {
    int ri = task / 17, tile = task % 17;   // ri=0 -> row t-1, ri=1 -> row t
    int f0 = -2 + 16*tile;
    int ax = ri*FW + PADF + f0 + m - 2;
    float bias = biasZR[m];
    v8f c;
#pragma unroll
    for (int j = 0; j < 8; ++j) c[j] = bias;
    for (int chk = 0; chk < 48; ++chk) {
      const float* p = &S[OFFzr[2*chk + g] + ax];
      v2f a; a.x = p[0]; a.y = p[1];
      v2f bb = Bzr2[chk*32 + lane];
      c = __builtin_amdgcn_wmma_f32_16x16x4_f32(false, a, false, bb, (short)0, c, false, false);
    }
    int fb = PADF + f0 + 8*g;
    if (m < 8) {
      if (ri == 1) {
#pragma unroll
        for (int j = 0; j < 8; ++j)
          S[ZBASE + m*FW + fb + j] = fast_sigmoid(c[j]);
      }
    } else {
      int chn  = m - 8;
      int hrow = HBASE + (chn*3 + ri + 1)*FW;
      int rrow = RHBASE + (chn*2 + ri)*FW;
#pragma unroll
      for (int j = 0; j < 8; ++j) {
        float rr = fast_sigmoid(c[j]);
        S[rrow + fb + j] = rr * S[hrow + fb + j];
      }
    }
  }
  __syncthreads();

  // ---- phase 3: n = tanh(conv(xp,xn)+conv(rh,hn)); out = (1-z)h + z n ----
  for (int task = wave; task < 17; task += 8) {
    int f0 = 16*task;
    int ax = PADF + f0 + m - 2;
    float bias = biasN[m];
    v8f c;
#pragma unroll
    for (int j = 0; j < 8; ++j) c[j] = bias;
    for (int chk = 0; chk < 48; ++chk) {
      const float* p = &S[OFFn[2*chk + g] + ax];
      v2f a; a.x = p[0]; a.y = p[1];
      v2f bb = Bn2[chk*32 + lane];
      c = __builtin_amdgcn_wmma_f32_16x16x4_f32(false, a, false, bb, (short)0, c, false, false);
    }
    if (m < 8) {
#pragma unroll
      for (int j = 0; j < 8; ++j) {
        int f = f0 + 8*g + j;
        if (f < NF) {
          float nn = fast_tanh(c[j]);
          float z  = S[ZBASE + m*FW + PADF + f];
          float hh = S[HBASE + (m*3 + 2)*FW + PADF + f];
          out[((b*8 + m)*NT + t)*NF + f] = (1.f - z)*hh + z*nn;
        }
      }
    }
  }
}

extern "C" void kernel_launch(void* const* d_in, const int* in_sizes, int n_in,
                              void* d_out, int out_size, void* d_ws, size_t ws_size,
                              hipStream_t stream) {
  (void)in_sizes; (void)n_in; (void)out_size; (void)d_ws; (void)ws_size;
  dim3 grid(NT, NB), block(256);
  hipLaunchKernelGGL(ministage_gru_fused, grid, block, 0, stream,
      (const float*)d_in[0],  (const float*)d_in[1],
      (const float*)d_in[2],  (const float*)d_in[3],
      (const float*)d_in[4],  (const float*)d_in[5],
      (const float*)d_in[6],  (const float*)d_in[7],
      (const float*)d_in[8],  (const float*)d_in[9],
      (const float*)d_in[10], (const float*)d_in[11],
      (const float*)d_in[12], (const float*)d_in[13],
      (const float*)d_in[14], (const float*)d_in[15],
      (float*)d_out);
}